// RRWM_51118700757118
// MI455X (gfx1250) — compile-verified
//
#include <hip/hip_runtime.h>
#include <math.h>

typedef __attribute__((ext_vector_type(2))) float v2f;
typedef __attribute__((ext_vector_type(8))) float v8f;

#define ALPHA_C   0.2f
#define BETA_C    30.0f
#define SK_EPS_C  1e-4f
#define MAX_ITER_C 50
#define SK_ITER_C  20
#define LDSN 65   // LDS row stride (supports n_src,n_tgt up to 64)

// ---------------------------------------------------------------------------
// deg[b,i] = sum_j M[b,i,j]   (one wave32 per row, coalesced 128B reads)
// ---------------------------------------------------------------------------
__global__ __launch_bounds__(256) void rowsum_kernel(const float* __restrict__ M,
                                                     float* __restrict__ deg,
                                                     int mn, int rows_total) {
  int gw   = (blockIdx.x * blockDim.x + threadIdx.x) >> 5;
  int lane = threadIdx.x & 31;
  if (gw >= rows_total) return;
  const float* row = M + (size_t)gw * (size_t)mn;
  float s = 0.f;
  for (int j = lane; j < mn; j += 32) s += row[j];
  for (int off = 16; off > 0; off >>= 1) s += __shfl_down(s, off, 32);
  if (lane == 0) deg[gw] = s;
}

// ---------------------------------------------------------------------------
// c[b] = max_i deg[b,i] + (global min deg)*1e-5   (single block)
// ---------------------------------------------------------------------------
__global__ __launch_bounds__(1024) void scalars_kernel(const float* __restrict__ deg,
                                                       float* __restrict__ csc,
                                                       int mn, int B) {
  __shared__ float red[1024];
  int t = threadIdx.x;
  int total = B * mn;
  float mnv = 3.4e38f;
  for (int i = t; i < total; i += 1024) mnv = fminf(mnv, deg[i]);
  red[t] = mnv; __syncthreads();
  for (int s = 512; s > 0; s >>= 1) { if (t < s) red[t] = fminf(red[t], red[t + s]); __syncthreads(); }
  float dmin = red[0]; __syncthreads();

  int b = t >> 4, sub = t & 15;                 // 16 threads per batch
  float mx = -3.4e38f;
  if (b < B) for (int i = sub; i < mn; i += 16) mx = fmaxf(mx, deg[b * mn + i]);
  for (int off = 8; off > 0; off >>= 1) mx = fmaxf(mx, __shfl_down(mx, off, 16));
  if (b < B && sub == 0) csc[b] = mx + dmin * 1e-5f;
}

// ---------------------------------------------------------------------------
// v0[b, ti*n_src+si] = valid ? 1/(ns_src*ns_tgt) : 0 ;  flags init
// ---------------------------------------------------------------------------
__global__ __launch_bounds__(256) void init_kernel(float* __restrict__ V, int* __restrict__ flags,
                                                   const int* __restrict__ pnsrc,
                                                   const int* __restrict__ ns_src,
                                                   const int* __restrict__ ns_tgt,
                                                   int mn, int B) {
  int i = blockIdx.x * blockDim.x + threadIdx.x;
  if (i == 0) flags[0] = 0;
  if (i >= B * mn) return;
  int b = i / mn, j = i % mn;
  int n_src = pnsrc[0];
  int si = j % n_src, ti = j / n_src;
  bool valid = (si < ns_src[b]) && (ti < ns_tgt[b]);
  V[i] = valid ? 1.0f / (float)(ns_src[b] * ns_tgt[b]) : 0.f;
}

// ---------------------------------------------------------------------------
// BV = M @ V (raw, per batch) using V_WMMA_F32_16X16X4_F32.
// A = 16x4 tile of M (lanes 0-15: K{0,1}, lanes 16-31: K{2,3} per ISA layout).
// B = v[j..j+3] broadcast across all 16 columns -> every D column identical,
// so only B's K-lane mapping matters (same float2 pattern as A).
// Output from C/D layout: lane 0 acc[k] = row k, lane 16 acc[k] = row k+8.
// ---------------------------------------------------------------------------
__global__ __launch_bounds__(256) void matvec_wmma_kernel(const float* __restrict__ M,
                                                          const float* __restrict__ V,
                                                          float* __restrict__ BV,
                                                          int mn) {
  int wave = threadIdx.x >> 5;
  int lane = threadIdx.x & 31;
  int ntiles = mn >> 4;
  int tile = blockIdx.x * 8 + wave;
  if (tile >= ntiles) return;                  // uniform per-wave exit, EXEC stays all-ones
  int b  = blockIdx.y;
  int i0 = tile << 4;
  int r    = lane & 15;
  int koff = (lane >> 4) << 1;                 // 0 for lanes 0-15, 2 for lanes 16-31

  const float* Ap = M + ((size_t)b * (size_t)mn + (size_t)(i0 + r)) * (size_t)mn + koff;
  const float* Vp = V + (size_t)b * (size_t)mn + koff;

  v8f acc = {0.f, 0.f, 0.f, 0.f, 0.f, 0.f, 0.f, 0.f};
#pragma unroll 8
  for (int j = 0; j < mn; j += 4) {
    v2f a  = *(const v2f*)(Ap + j);            // M[i0+r, j+koff .. j+koff+1]
    v2f bb = *(const v2f*)(Vp + j);            // v[j+koff .. j+koff+1] (broadcast over N)
    acc = __builtin_amdgcn_wmma_f32_16x16x4_f32(false, a, false, bb,
                                                (short)0, acc, false, false);
  }
  if (r == 0) {                                // lanes 0 and 16 write rows 0-7 / 8-15
    size_t base = (size_t)b * (size_t)mn + (size_t)i0 + (size_t)((lane >> 4) << 3);
#pragma unroll
    for (int k = 0; k < 8; ++k) BV[base + k] = acc[k];
  }
}

// ---------------------------------------------------------------------------
// Per-batch tail: normalize, exp, masked Sinkhorn (20 iters, in LDS),
// blend, normalize, per-batch diff partial. One 256-thread block per batch.
// ---------------------------------------------------------------------------
__global__ __launch_bounds__(256) void post_kernel(const float* __restrict__ BV,
                                                   const float* __restrict__ csc,
                                                   const int* __restrict__ pnsrc,
                                                   const int* __restrict__ ns_src_a,
                                                   const int* __restrict__ ns_tgt_a,
                                                   float* __restrict__ V2,
                                                   float* __restrict__ diffpart,
                                                   int mn) {
  __shared__ float sdata[64 * LDSN];
  __shared__ float red[256];
  __shared__ float lsum[64];

  int b = blockIdx.x, t = threadIdx.x;
  int n_src = pnsrc[0];
  int n_tgt = mn / n_src;
  int nss = ns_src_a[b], nst = ns_tgt_a[b];
  int per = (mn + 255) >> 8;

  float bvloc[16];
  float asum = 0.f, amax = -3.4e38f;
  for (int k = 0; k < per; ++k) {
    int j = t + (k << 8);
    float x = 0.f;
    if (j < mn) { x = BV[(size_t)b * mn + j]; asum += fabsf(x); amax = fmaxf(amax, x); }
    bvloc[k] = x;
  }
  red[t] = asum; __syncthreads();
  for (int s = 128; s > 0; s >>= 1) { if (t < s) red[t] += red[t + s]; __syncthreads(); }
  asum = red[0]; __syncthreads();
  red[t] = amax; __syncthreads();
  for (int s = 128; s > 0; s >>= 1) { if (t < s) red[t] = fmaxf(red[t], red[t + s]); __syncthreads(); }
  amax = red[0]; __syncthreads();

  float inv = 1.0f / asum;             // v1 = raw*inv (per-batch scalar c cancels)
  float scl = BETA_C / amax;           // BETA*v1/max(v1) == BETA*raw/max(raw)

  for (int k = 0; k < per; ++k) {
    int j = t + (k << 8);
    if (j < mn) {
      int si = j % n_src, ti = j / n_src;
      float sv = expf(scl * bvloc[k]);
      bool valid = (si < nss) && (ti < nst);
      sdata[si * LDSN + ti] = valid ? (sv + SK_EPS_C) : 0.f;
    }
  }
  __syncthreads();

  for (int it = 0; it < SK_ITER_C; ++it) {
    if ((it & 1) == 0) {               // row-normalize (sum over ti)
      if (t < n_src) {
        float rs = 0.f;
        for (int ti = 0; ti < n_tgt; ++ti) rs += sdata[t * LDSN + ti];
        lsum[t] = (rs > 0.f) ? rs : 1.0f;
      }
      __syncthreads();
      for (int k = 0; k < per; ++k) {
        int j = t + (k << 8);
        if (j < mn) { int si = j % n_src, ti = j / n_src; sdata[si * LDSN + ti] /= lsum[si]; }
      }
    } else {                           // col-normalize (sum over si)
      if (t < n_tgt) {
        float cs = 0.f;
        for (int si = 0; si < n_src; ++si) cs += sdata[si * LDSN + t];
        lsum[t] = (cs > 0.f) ? cs : 1.0f;
      }
      __syncthreads();
      for (int k = 0; k < per; ++k) {
        int j = t + (k << 8);
        if (j < mn) { int si = j % n_src, ti = j / n_src; sdata[si * LDSN + ti] /= lsum[ti]; }
      }
    }
    __syncthreads();
  }

  float v2loc[16];
  float asum2 = 0.f;
  for (int k = 0; k < per; ++k) {
    int j = t + (k << 8);
    float v2v = 0.f;
    if (j < mn) {
      int si = j % n_src, ti = j / n_src;
      float v1v = bvloc[k] * inv;
      v2v = ALPHA_C * sdata[si * LDSN + ti] + (1.0f - ALPHA_C) * v1v;
      asum2 += fabsf(v2v);
    }
    v2loc[k] = v2v;
  }
  red[t] = asum2; __syncthreads();
  for (int s = 128; s > 0; s >>= 1) { if (t < s) red[t] += red[t + s]; __syncthreads(); }
  asum2 = red[0]; __syncthreads();

  float inv2 = 1.0f / asum2;
  float invc = 1.0f / csc[b];          // last_v = raw / c[b]
  float dsq = 0.f;
  for (int k = 0; k < per; ++k) {
    int j = t + (k << 8);
    if (j < mn) {
      float v2n = v2loc[k] * inv2;
      V2[(size_t)b * mn + j] = v2n;
      float dd = v2n - bvloc[k] * invc;
      dsq += dd * dd;
    }
  }
  red[t] = dsq; __syncthreads();
  for (int s = 128; s > 0; s >>= 1) { if (t < s) red[t] += red[t + s]; __syncthreads(); }
  if (t == 0) diffpart[b] = red[0];
}

// ---------------------------------------------------------------------------
// V = done(old) ? V : V2   (runs BEFORE flag update, matching reference order)
// ---------------------------------------------------------------------------
__global__ __launch_bounds__(256) void select_kernel(float* __restrict__ V,
                                                     const float* __restrict__ V2,
                                                     const int* __restrict__ flags,
                                                     int total) {
  int i = blockIdx.x * blockDim.x + threadIdx.x;
  if (i < total && flags[0] == 0) V[i] = V2[i];
}

// done |= (||v2 - last_v||_F < 1e-5); deterministic serial sum of B partials
__global__ void finalize_kernel(const float* __restrict__ diffpart,
                                int* __restrict__ flags, int B) {
  if (threadIdx.x == 0 && blockIdx.x == 0) {
    float s = 0.f;
    for (int b = 0; b < B; ++b) s += diffpart[b];
    if (sqrtf(s) < 1e-5f) flags[0] = 1;
  }
}

// ---------------------------------------------------------------------------
extern "C" void kernel_launch(void* const* d_in, const int* in_sizes, int n_in,
                              void* d_out, int out_size, void* d_ws, size_t ws_size,
                              hipStream_t stream) {
  const float* M      = (const float*)d_in[0];
  const int*   pnsrc  = (const int*)d_in[1];   // num_src (device scalar)
  const int*   ns_src = (const int*)d_in[2];
  const int*   ns_tgt = (const int*)d_in[3];

  int B  = in_sizes[2];                 // 64
  int mn = out_size / B;                // 2304
  size_t Bmn = (size_t)B * (size_t)mn;

  float* wsf      = (float*)d_ws;
  float* deg      = wsf;                // B*mn
  float* csc      = deg + Bmn;          // B
  float* V        = csc + B;            // B*mn
  float* V2       = V + Bmn;            // B*mn
  float* BV       = V2 + Bmn;           // B*mn
  float* diffpart = BV + Bmn;           // B
  int*   flags    = (int*)(diffpart + B);

  int rows = B * mn;
  rowsum_kernel<<<(rows + 7) / 8, 256, 0, stream>>>(M, deg, mn, rows);
  scalars_kernel<<<1, 1024, 0, stream>>>(deg, csc, mn, B);
  init_kernel<<<(int)((Bmn + 255) / 256), 256, 0, stream>>>(V, flags, pnsrc, ns_src, ns_tgt, mn, B);

  int ntiles = mn >> 4;
  dim3 mvGrid((ntiles + 7) / 8, B);
  for (int it = 0; it < MAX_ITER_C; ++it) {
    matvec_wmma_kernel<<<mvGrid, 256, 0, stream>>>(M, V, BV, mn);
    post_kernel<<<B, 256, 0, stream>>>(BV, csc, pnsrc, ns_src, ns_tgt, V2, diffpart, mn);
    select_kernel<<<(int)((Bmn + 255) / 256), 256, 0, stream>>>(V, V2, flags, (int)Bmn);
    finalize_kernel<<<1, 32, 0, stream>>>(diffpart, flags, B);
  }
  hipMemcpyAsync(d_out, V, Bmn * sizeof(float), hipMemcpyDeviceToDevice, stream);
}